// TopologyRobustLocalAttention_5446018531697
// MI455X (gfx1250) — compile-verified
//
#include <hip/hip_runtime.h>
#include <hip/hip_bf16.h>

// Problem constants (match reference).
#define NU 100000
#define NI 50000
#define NEDGE 500000
#define H 64

typedef __attribute__((ext_vector_type(2))) float v2f;
typedef __attribute__((ext_vector_type(8))) float v8f;

// ---------------------------------------------------------------------------
// Wave-level 16x64 = 16x64 @ 64x64 (+bias) GEMM using V_WMMA_F32_16X16X4_F32.
// `in`/`out` are generic pointers (LDS or global), row-major stride 64.
// A layout (16x4 f32): lane m = lane&15; VGPR pair holds K = (lane>>4)*2 +{0,1}
// B layout (4x16 f32): lane n = lane&15; VGPR pair holds K = (lane>>4)*2 +{0,1}
// C/D layout (16x16 f32): vgpr r -> row r + (lane>>4)*8, col = lane&15
// ---------------------------------------------------------------------------
__device__ __forceinline__ void wave_gemm_16x64(const float* __restrict__ in,
                                                const float* __restrict__ W,
                                                const float* __restrict__ bias,
                                                float* __restrict__ out) {
  const int lane  = threadIdx.x & 31;
  const int m     = lane & 15;          // A row / D col-index
  const int koff  = (lane >> 4) * 2;    // which K pair this half-wave owns
  const int rbase = (lane >> 4) * 8;    // D row base

#pragma unroll
  for (int j = 0; j < 4; ++j) {
    const int col = j * 16 + m;
    const float bval = bias ? bias[col] : 0.0f;
    v8f c;
#pragma unroll
    for (int r = 0; r < 8; ++r) c[r] = bval;

#pragma unroll
    for (int kc = 0; kc < 16; ++kc) {
      const int k0 = kc * 4 + koff;
      v2f a = *(const v2f*)(in + m * 64 + k0);   // 8B-aligned pair
      v2f b;
      b.x = W[(k0 + 0) * 64 + col];
      b.y = W[(k0 + 1) * 64 + col];
      // (neg_a, A, neg_b, B, c_mod, C, reuse_a, reuse_b)
      c = __builtin_amdgcn_wmma_f32_16x16x4_f32(false, a, false, b,
                                                (short)0, c, false, false);
    }
#pragma unroll
    for (int r = 0; r < 8; ++r) out[(rbase + r) * 64 + col] = c[r];
  }
}

// ---------------------------------------------------------------------------
// Fused node phase: per 16-row strip of nodes, compute
//   h  = x @ Win + bin                (LDS)
//   k  = h @ Wk  + bk                 (LDS)
//   Ak = k @ WaTop                    (global)
//   v  = h @ Wv  + bv                 (global)
//   q  = h @ Wq  + bq                 (LDS)
//   Aq = q @ WaBot + ba               (global)
// 5 waves per block, one strip per wave; grid sized exactly (no tails).
// ---------------------------------------------------------------------------
#define NPW 5  // waves (strips) per block
__global__ void node_phase_kernel(const float* __restrict__ x,
                                  const float* __restrict__ Win, const float* __restrict__ bin,
                                  const float* __restrict__ Wk,  const float* __restrict__ bk,
                                  const float* __restrict__ WaTop,
                                  const float* __restrict__ Wv,  const float* __restrict__ bv,
                                  const float* __restrict__ Wq,  const float* __restrict__ bq,
                                  const float* __restrict__ WaBot, const float* __restrict__ ba,
                                  float* __restrict__ Ak, float* __restrict__ Vv,
                                  float* __restrict__ Aq) {
  __shared__ float bufH[NPW][16 * 64];
  __shared__ float bufT[NPW][16 * 64];
  const int    wave = threadIdx.x >> 5;
  const size_t strip = (size_t)blockIdx.x * NPW + wave;
  const size_t row0  = strip * 16;

  float* h = bufH[wave];
  float* t = bufT[wave];

  wave_gemm_16x64(x + row0 * 64, Win, bin, h);     // h = x@Win + bin
  __syncthreads();
  wave_gemm_16x64(h, Wk, bk, t);                   // k = h@Wk + bk
  __syncthreads();
  wave_gemm_16x64(t, WaTop, nullptr, Ak + row0 * 64);  // Ak = k@WaTop
  wave_gemm_16x64(h, Wv, bv, Vv + row0 * 64);          // v  = h@Wv + bv
  __syncthreads();
  wave_gemm_16x64(h, Wq, bq, t);                   // q = h@Wq + bq
  __syncthreads();
  wave_gemm_16x64(t, WaBot, ba, Aq + row0 * 64);   // Aq = q@WaBot + ba
}

// ---------------------------------------------------------------------------
// Edge phase: one wave per edge iteration; lane owns channels {2*lane, 2*lane+1}.
//   m = sigmoid((Ak[src]+Aq[dst]) * hw) * v[src];  agg[dst] += m  (f32 atomics)
// ---------------------------------------------------------------------------
__global__ void edge_kernel(const int* __restrict__ src, const int* __restrict__ dst,
                            int nedges,
                            const float* __restrict__ Ak, const float* __restrict__ Aq,
                            const float* __restrict__ Vv, const float* __restrict__ hw,
                            float* __restrict__ agg) {
  const int lane = threadIdx.x & 31;
  const int wid  = (blockIdx.x * blockDim.x + threadIdx.x) >> 5;
  const int nw   = (gridDim.x * blockDim.x) >> 5;
  const float2 hw2 = *(const float2*)(hw + 2 * lane);

  for (int e = wid; e < nedges; e += nw) {
    const size_t s = (size_t)src[e];
    const size_t d = (size_t)dst[e];
    const float2 aS = *(const float2*)(Ak + s * 64 + 2 * lane);
    const float2 aD = *(const float2*)(Aq + d * 64 + 2 * lane);
    const float2 v  = *(const float2*)(Vv + s * 64 + 2 * lane);
    const float ax = (aS.x + aD.x) * hw2.x;
    const float ay = (aS.y + aD.y) * hw2.y;
    const float mx = v.x / (1.0f + __expf(-ax));
    const float my = v.y / (1.0f + __expf(-ay));
    float* p = agg + d * 64 + 2 * lane;
    __hip_atomic_fetch_add(p + 0, mx, __ATOMIC_RELAXED, __HIP_MEMORY_SCOPE_AGENT);
    __hip_atomic_fetch_add(p + 1, my, __ATOMIC_RELAXED, __HIP_MEMORY_SCOPE_AGENT);
  }
}

// ---------------------------------------------------------------------------
// Output phase: out[strip] = agg[strip] @ Wout + bout, one strip per wave.
// ---------------------------------------------------------------------------
__global__ void out_gemm_kernel(const float* __restrict__ in,
                                const float* __restrict__ W,
                                const float* __restrict__ bias,
                                float* __restrict__ out, int nstrips) {
  const int wid = (blockIdx.x * blockDim.x + threadIdx.x) >> 5;
  const int nw  = (gridDim.x * blockDim.x) >> 5;
  for (int s = wid; s < nstrips; s += nw) {
    wave_gemm_16x64(in + (size_t)s * 16 * 64, W, bias, out + (size_t)s * 16 * 64);
  }
}

__global__ void zero_kernel(float4* __restrict__ p, size_t n4) {
  const size_t i0 = (size_t)blockIdx.x * blockDim.x + threadIdx.x;
  const size_t stride = (size_t)gridDim.x * blockDim.x;
  const float4 z = {0.f, 0.f, 0.f, 0.f};
  for (size_t i = i0; i < n4; i += stride) p[i] = z;
}

// ---------------------------------------------------------------------------
extern "C" void kernel_launch(void* const* d_in, const int* in_sizes, int n_in,
                              void* d_out, int out_size, void* d_ws, size_t ws_size,
                              hipStream_t stream) {
  // Inputs in setup_inputs() order
  const float* x_user = (const float*)d_in[0];
  const float* x_item = (const float*)d_in[1];
  const int*   src_ui = (const int*)d_in[2];
  const int*   dst_ui = (const int*)d_in[3];
  const int*   src_iu = (const int*)d_in[4];
  const int*   dst_iu = (const int*)d_in[5];
  const float* Win_u = (const float*)d_in[6];   const float* bin_u = (const float*)d_in[7];
  const float* Win_i = (const float*)d_in[8];   const float* bin_i = (const float*)d_in[9];
  const float* Wout_u = (const float*)d_in[10]; const float* bout_u = (const float*)d_in[11];
  const float* Wout_i = (const float*)d_in[12]; const float* bout_i = (const float*)d_in[13];
  const float* Wk_ui = (const float*)d_in[14];  const float* bk_ui = (const float*)d_in[15];
  const float* Wq_ui = (const float*)d_in[16];  const float* bq_ui = (const float*)d_in[17];
  const float* Wv_ui = (const float*)d_in[18];  const float* bv_ui = (const float*)d_in[19];
  const float* Wa_ui = (const float*)d_in[20];  const float* ba_ui = (const float*)d_in[21];
  const float* hw_ui = (const float*)d_in[22];
  const float* Wk_iu = (const float*)d_in[23];  const float* bk_iu = (const float*)d_in[24];
  const float* Wq_iu = (const float*)d_in[25];  const float* bq_iu = (const float*)d_in[26];
  const float* Wv_iu = (const float*)d_in[27];  const float* bv_iu = (const float*)d_in[28];
  const float* Wa_iu = (const float*)d_in[29];  const float* ba_iu = (const float*)d_in[30];
  const float* hw_iu = (const float*)d_in[31];

  const int nE = in_sizes[2];

  // Wa is [128,64] row-major: top half multiplies k (src side), bottom half q (dst side).
  const float* WaTop_ui = Wa_ui;            const float* WaBot_ui = Wa_ui + 64 * 64;
  const float* WaTop_iu = Wa_iu;            const float* WaBot_iu = Wa_iu + 64 * 64;

  // Workspace layout (floats): Ak_u, Aq_u, V_u, Ak_i, Aq_i, V_i, agg_u, agg_i
  const size_t NU64 = (size_t)NU * 64;
  const size_t NI64 = (size_t)NI * 64;
  float* ws   = (float*)d_ws;
  float* Ak_u = ws;
  float* Aq_u = Ak_u + NU64;
  float* V_u  = Aq_u + NU64;
  float* Ak_i = V_u + NU64;
  float* Aq_i = Ak_i + NI64;
  float* V_i  = Aq_i + NI64;
  float* agg_u = V_i + NI64;   // agg_u and agg_i contiguous -> one zero pass
  float* agg_i = agg_u + NU64;

  // 0) zero the aggregation buffers (must happen every call: graph replays).
  zero_kernel<<<2048, 256, 0, stream>>>((float4*)agg_u, (NU64 + NI64) / 4);

  // 1) node phase (fused 6-GEMM chain, WMMA f32).
  //    users: Ak/V for ui-edges (src side), Aq for iu-edges (dst side, carries ba_iu)
  node_phase_kernel<<<NU / (16 * NPW), 32 * NPW, 0, stream>>>(
      x_user, Win_u, bin_u,
      Wk_ui, bk_ui, WaTop_ui,
      Wv_ui, bv_ui,
      Wq_iu, bq_iu, WaBot_iu, ba_iu,
      Ak_u, V_u, Aq_u);
  //    items: Ak/V for iu-edges (src side), Aq for ui-edges (dst side, carries ba_ui)
  node_phase_kernel<<<NI / (16 * NPW), 32 * NPW, 0, stream>>>(
      x_item, Win_i, bin_i,
      Wk_iu, bk_iu, WaTop_iu,
      Wv_iu, bv_iu,
      Wq_ui, bq_ui, WaBot_ui, ba_ui,
      Ak_i, V_i, Aq_i);

  // 2) edge phase (gather + gate + f32 atomic scatter-add).
  edge_kernel<<<2048, 256, 0, stream>>>(src_ui, dst_ui, nE, Ak_u, Aq_i, V_u, hw_ui, agg_i);
  edge_kernel<<<2048, 256, 0, stream>>>(src_iu, dst_iu, nE, Ak_i, Aq_u, V_i, hw_iu, agg_u);

  // 3) output linears straight into d_out (out_u first, then out_i).
  float* out = (float*)d_out;
  out_gemm_kernel<<<800, 256, 0, stream>>>(agg_u, Wout_u, bout_u, out, NU / 16);
  out_gemm_kernel<<<400, 256, 0, stream>>>(agg_i, Wout_i, bout_i, out + NU64, NI / 16);
}